// SASRec_88158498718039
// MI455X (gfx1250) — compile-verified
//
#include <hip/hip_runtime.h>
#include <hip/hip_bf16.h>
#include <math.h>

// ---- problem constants (match reference) ----
#define NUM_ITEMS 50000
#define DMODEL    64
#define NLAYER    2
#define BB        128
#define SS        50
#define MROWS     (BB * SS)          // 6400 rows, = 400 * 16
#define BSD       (MROWS * DMODEL)   // 409600 floats per activation buffer
#define LN_EPS    1e-5f

typedef float v2f __attribute__((ext_vector_type(2)));
typedef float v8f __attribute__((ext_vector_type(8)));

// -------------------------------------------------------------------------
// x[b,s,:] = emb[item_ids[b,s]] + pos_emb[s]
// -------------------------------------------------------------------------
__global__ __launch_bounds__(256) void embed_kernel(
    const int* __restrict__ ids, const float* __restrict__ emb,
    const float* __restrict__ pos, float* __restrict__ X)
{
    int idx = blockIdx.x * 256 + threadIdx.x;        // 0 .. BSD-1
    int d   = idx & (DMODEL - 1);
    int row = idx >> 6;                              // b*S + s
    int s   = row % SS;
    int it  = ids[row];
    X[idx] = emb[it * DMODEL + d] + pos[s * DMODEL + d];
}

// -------------------------------------------------------------------------
// Y[m, n0+n] = sum_k X[m,k] * W[n0+n, k]  (+ bias, optional relu)
// One wave computes a 16x16 tile via V_WMMA_F32_16X16X4_F32.
// fp32 WMMA layout (ISA 7.12.2):
//   A 16x4 : lane -> M = lane%16, VGPR0/1 hold K = koff, koff+1 (koff = (lane<16)?0:2)
//   B 4x16 : lane -> N = lane%16, VGPR0/1 hold K = koff, koff+1
//   C 16x16: VGPR r -> M = r + (lane<16?0:8), N = lane%16
// -------------------------------------------------------------------------
__global__ __launch_bounds__(32) void gemm64_kernel(
    const float* __restrict__ X, const float* __restrict__ W,
    const float* __restrict__ bias, float* __restrict__ Y, int relu)
{
    const int m0   = blockIdx.x * 16;
    const int n0   = blockIdx.y * 16;
    const int lane = threadIdx.x;
    const int half = lane >> 4;       // 0 | 1
    const int l16  = lane & 15;
    const int koff = half * 2;

    const float* arow = X + (m0 + l16) * DMODEL;  // row m, contiguous K
    const float* brow = W + (n0 + l16) * DMODEL;  // W[n,:] : B[k][n] = W[n][k]

    v8f c = {};
#pragma unroll
    for (int kk = 0; kk < 16; ++kk) {             // K = 64 in steps of 4
        v2f a = *(const v2f*)(arow + kk * 4 + koff);
        v2f b = *(const v2f*)(brow + kk * 4 + koff);
        c = __builtin_amdgcn_wmma_f32_16x16x4_f32(
                false, a, false, b, (short)0, c, false, false);
    }

    const int n = n0 + l16;
    const float bv = bias ? bias[n] : 0.0f;
#pragma unroll
    for (int r = 0; r < 8; ++r) {
        int m = m0 + r + half * 8;
        float v = c[r] + bv;
        if (relu) v = fmaxf(v, 0.0f);
        Y[m * DMODEL + n] = v;
    }
}

// -------------------------------------------------------------------------
// Causal attention, one workgroup per batch element. K,V staged in LDS.
// One wave per query row; lanes parallel over keys then over features.
// -------------------------------------------------------------------------
__global__ __launch_bounds__(256) void attention_kernel(
    const float* __restrict__ Q, const float* __restrict__ K,
    const float* __restrict__ V, float* __restrict__ O)
{
    __shared__ float sK[SS * DMODEL];
    __shared__ float sV[SS * DMODEL];

    const int b = blockIdx.x;
    const float* Kb = K + b * SS * DMODEL;
    const float* Vb = V + b * SS * DMODEL;
    for (int i = threadIdx.x; i < SS * DMODEL; i += 256) {
        sK[i] = Kb[i];
        sV[i] = Vb[i];
    }
    __syncthreads();

    const int wave = threadIdx.x >> 5;
    const int lane = threadIdx.x & 31;
    const float scale = 0.125f;                    // 1/sqrt(64)

    for (int q = wave; q < SS; q += 8) {
        const float* qr = Q + (b * SS + q) * DMODEL;
        const int k0 = lane, k1 = lane + 32;

        float s0 = -INFINITY, s1 = -INFINITY;
        if (k0 <= q) {
            float acc = 0.f;
            for (int d = 0; d < DMODEL; ++d) acc += qr[d] * sK[k0 * DMODEL + d];
            s0 = acc * scale;
        }
        if (k1 <= q && k1 < SS) {
            float acc = 0.f;
            for (int d = 0; d < DMODEL; ++d) acc += qr[d] * sK[k1 * DMODEL + d];
            s1 = acc * scale;
        }

        float mx = fmaxf(s0, s1);
        for (int off = 16; off > 0; off >>= 1)
            mx = fmaxf(mx, __shfl_xor(mx, off, 32));

        float e0 = (k0 <= q)             ? __expf(s0 - mx) : 0.f;
        float e1 = (k1 <= q && k1 < SS)  ? __expf(s1 - mx) : 0.f;
        float sum = e0 + e1;
        for (int off = 16; off > 0; off >>= 1)
            sum += __shfl_xor(sum, off, 32);
        const float inv = 1.0f / sum;

        float o0 = 0.f, o1 = 0.f;
        for (int k = 0; k <= q; ++k) {
            float p = (k < 32) ? __shfl(e0, k, 32) : __shfl(e1, k - 32, 32);
            o0 += p * sV[k * DMODEL + lane];
            o1 += p * sV[k * DMODEL + lane + 32];
        }
        float* orow = O + (b * SS + q) * DMODEL;
        orow[lane]      = o0 * inv;
        orow[lane + 32] = o1 * inv;
    }
}

// -------------------------------------------------------------------------
// X[row] = LayerNorm(X[row] + R[row]; g, b), in place. One wave per row.
// -------------------------------------------------------------------------
__global__ __launch_bounds__(256) void add_ln_kernel(
    float* __restrict__ X, const float* __restrict__ R,
    const float* __restrict__ g, const float* __restrict__ bta)
{
    const int row  = blockIdx.x * 8 + (threadIdx.x >> 5);
    const int lane = threadIdx.x & 31;

    float* xr       = X + row * DMODEL;
    const float* rr = R + row * DMODEL;

    float v0 = xr[lane]      + rr[lane];
    float v1 = xr[lane + 32] + rr[lane + 32];

    float s = v0 + v1;
    for (int off = 16; off > 0; off >>= 1) s += __shfl_xor(s, off, 32);
    const float mu = s * (1.0f / 64.0f);

    const float d0 = v0 - mu, d1 = v1 - mu;
    float vs = d0 * d0 + d1 * d1;
    for (int off = 16; off > 0; off >>= 1) vs += __shfl_xor(vs, off, 32);
    const float rstd = rsqrtf(vs * (1.0f / 64.0f) + LN_EPS);

    xr[lane]      = g[lane]      * d0 * rstd + bta[lane];
    xr[lane + 32] = g[lane + 32] * d1 * rstd + bta[lane + 32];
}

// -------------------------------------------------------------------------
// logits[m, v] = sum_d X[m,d] * E[v,d]   ->  [6400, 50000] fp32
// HBM-store-bound (1.28 GB >> 192 MB L2) -> non-temporal stores.
// Each wave computes 5 n-tiles (16x80 macro-tile): one A float2 feeds
// 5 WMMAs per K-step, cutting A-side L0/L2 traffic 5x; the 5 B tiles are
// 4 KB apart -> single base + immediate offsets.
// block = 128 (4 waves = 4 m-tiles), grid = (50000/80, 6400/64) = (625, 100).
// -------------------------------------------------------------------------
__global__ __launch_bounds__(128) void logits_kernel(
    const float* __restrict__ X, const float* __restrict__ E,
    float* __restrict__ out)
{
    const int wave = threadIdx.x >> 5;
    const int lane = threadIdx.x & 31;
    const int n0   = blockIdx.x * 80;              // 5 n-tiles of 16
    const int m0   = (blockIdx.y * 4 + wave) * 16;
    const int half = lane >> 4;
    const int l16  = lane & 15;
    const int koff = half * 2;

    const float* arow = X + (m0 + l16) * DMODEL;
    const float* brow = E + (n0 + l16) * DMODEL;   // B[k][n] = E[n][k]

    v8f c0 = {}, c1 = {}, c2 = {}, c3 = {}, c4 = {};
#pragma unroll
    for (int kk = 0; kk < 16; ++kk) {
        const int o = kk * 4 + koff;
        v2f a  = *(const v2f*)(arow + o);
        v2f b0 = *(const v2f*)(brow + o);
        v2f b1 = *(const v2f*)(brow + o + 16 * DMODEL);
        v2f b2 = *(const v2f*)(brow + o + 32 * DMODEL);
        v2f b3 = *(const v2f*)(brow + o + 48 * DMODEL);
        v2f b4 = *(const v2f*)(brow + o + 64 * DMODEL);
        c0 = __builtin_amdgcn_wmma_f32_16x16x4_f32(false, a, false, b0, (short)0, c0, false, false);
        c1 = __builtin_amdgcn_wmma_f32_16x16x4_f32(false, a, false, b1, (short)0, c1, false, false);
        c2 = __builtin_amdgcn_wmma_f32_16x16x4_f32(false, a, false, b2, (short)0, c2, false, false);
        c3 = __builtin_amdgcn_wmma_f32_16x16x4_f32(false, a, false, b3, (short)0, c3, false, false);
        c4 = __builtin_amdgcn_wmma_f32_16x16x4_f32(false, a, false, b4, (short)0, c4, false, false);
    }

    const int n = n0 + l16;
#pragma unroll
    for (int r = 0; r < 8; ++r) {
        size_t m = (size_t)(m0 + r + half * 8);
        float* orow = out + m * (size_t)NUM_ITEMS + n;
        __builtin_nontemporal_store(c0[r], orow);
        __builtin_nontemporal_store(c1[r], orow + 16);
        __builtin_nontemporal_store(c2[r], orow + 32);
        __builtin_nontemporal_store(c3[r], orow + 48);
        __builtin_nontemporal_store(c4[r], orow + 64);
    }
}

// -------------------------------------------------------------------------
// host-side orchestration
// -------------------------------------------------------------------------
extern "C" void kernel_launch(void* const* d_in, const int* in_sizes, int n_in,
                              void* d_out, int out_size, void* d_ws, size_t ws_size,
                              hipStream_t stream)
{
    const int*   ids  = (const int*)  d_in[0];
    const float* emb  = (const float*)d_in[1];
    const float* pos  = (const float*)d_in[2];
    const float* Wq   = (const float*)d_in[3];
    const float* Wk   = (const float*)d_in[4];
    const float* Wv   = (const float*)d_in[5];
    const float* W1   = (const float*)d_in[6];
    const float* b1   = (const float*)d_in[7];
    const float* W2   = (const float*)d_in[8];
    const float* b2   = (const float*)d_in[9];
    const float* lng  = (const float*)d_in[10];
    const float* lnb  = (const float*)d_in[11];
    float*       out  = (float*)d_out;

    float* ws = (float*)d_ws;
    float* X  = ws;              // activations, updated in place
    float* Qb = ws + 1 * BSD;    // Q, later reused for H
    float* Kb = ws + 2 * BSD;    // K, later reused for FF
    float* Vb = ws + 3 * BSD;    // V
    float* Tb = ws + 4 * BSD;    // attention output

    embed_kernel<<<BSD / 256, 256, 0, stream>>>(ids, emb, pos, X);

    const dim3 gg(MROWS / 16, DMODEL / 16);   // (400, 4)
    for (int l = 0; l < NLAYER; ++l) {
        const float* wq = Wq + l * DMODEL * DMODEL;
        const float* wk = Wk + l * DMODEL * DMODEL;
        const float* wv = Wv + l * DMODEL * DMODEL;
        const float* w1 = W1 + l * DMODEL * DMODEL;
        const float* w2 = W2 + l * DMODEL * DMODEL;
        const float* bb1 = b1 + l * DMODEL;
        const float* bb2 = b2 + l * DMODEL;
        const float* g   = lng + l * DMODEL;
        const float* bt  = lnb + l * DMODEL;

        gemm64_kernel<<<gg, 32, 0, stream>>>(X, wq, nullptr, Qb, 0);
        gemm64_kernel<<<gg, 32, 0, stream>>>(X, wk, nullptr, Kb, 0);
        gemm64_kernel<<<gg, 32, 0, stream>>>(X, wv, nullptr, Vb, 0);

        attention_kernel<<<BB, 256, 0, stream>>>(Qb, Kb, Vb, Tb);
        add_ln_kernel<<<MROWS / 8, 256, 0, stream>>>(X, Tb, g, bt);

        gemm64_kernel<<<gg, 32, 0, stream>>>(X, w1, bb1, Qb, 1);   // H = relu(...)
        gemm64_kernel<<<gg, 32, 0, stream>>>(Qb, w2, bb2, Kb, 0);  // FF
        add_ln_kernel<<<MROWS / 8, 256, 0, stream>>>(X, Kb, g, bt);
    }

    const dim3 gl(NUM_ITEMS / 80, MROWS / 64);  // (625, 100), 4 waves/block
    logits_kernel<<<gl, 128, 0, stream>>>(X, emb, out);
}